// OPTAttention_26345329393725
// MI455X (gfx1250) — compile-verified
//
#include <hip/hip_runtime.h>

typedef __attribute__((ext_vector_type(2))) float v2f;
typedef __attribute__((ext_vector_type(8))) float v8f;

constexpr int kB  = 32;     // batch
constexpr int kL  = 2048;   // context length per sequence
constexpr int kD  = 2048;   // model dim
constexpr int kH  = 32;     // heads
constexpr int kHD = 64;     // head dim
constexpr float kScale = 0.125f;  // 1/sqrt(64)

// ---------------------------------------------------------------------------
// fp32 WMMA helper: D = A(16x4) * B(4x16) + C(16x16), wave32
// A layout per ISA: lanes 0-15 -> M=0..15 {K0,K1}, lanes 16-31 -> M=0..15 {K2,K3}
// B layout (symmetric over N): lanes 0-15 -> N {K0,K1}, lanes 16-31 -> N {K2,K3}
// C/D: VGPR r, lane l -> M = r + 8*(l>>4), N = l&15
// ---------------------------------------------------------------------------
__device__ __forceinline__ v8f wmma4(v2f a, v2f b, v8f c) {
  return __builtin_amdgcn_wmma_f32_16x16x4_f32(
      /*neg_a=*/false, a, /*neg_b=*/false, b,
      /*c_mod=*/(short)0, c, /*reuse_a=*/false, /*reuse_b=*/false);
}

// ===========================================================================
// Kernel 1: q,k,v = h @ {Wq,Wk,Wv} + bias.  One wave per 32(M) x 16(N) tile.
// grid = (kD/16, 3), block = 32 (one wave, EXEC all ones for WMMA)
// ===========================================================================
__global__ __launch_bounds__(32)
void qkv_proj_kernel(const float* __restrict__ h,
                     const float* __restrict__ Wq, const float* __restrict__ bq,
                     const float* __restrict__ Wk, const float* __restrict__ bk,
                     const float* __restrict__ Wv, const float* __restrict__ bv,
                     float* __restrict__ q, float* __restrict__ k,
                     float* __restrict__ v) {
  const int ntile = blockIdx.x;              // 0..127
  const float* W; const float* bias; float* out;
  if (blockIdx.y == 0)      { W = Wq; bias = bq; out = q; }
  else if (blockIdx.y == 1) { W = Wk; bias = bk; out = k; }
  else                      { W = Wv; bias = bv; out = v; }

  const int lane = threadIdx.x & 31;
  const int m    = lane & 15;     // M row (A) / N col (B,D)
  const int half = lane >> 4;     // selects K pair {0,1} vs {2,3}
  const int ncol = ntile * 16 + m;

  v8f acc0 = {};  // rows 0..15 of h
  v8f acc1 = {};  // rows 16..31 of h
  for (int k0 = 0; k0 < kD; k0 += 4) {
    const int krow = k0 + 2 * half;
    v2f a0 = *(const v2f*)(h + (size_t)m        * kD + krow);
    v2f a1 = *(const v2f*)(h + (size_t)(16 + m) * kD + krow);
    v2f bm;
    bm.x = W[(size_t)krow       * kD + ncol];
    bm.y = W[(size_t)(krow + 1) * kD + ncol];
    acc0 = wmma4(a0, bm, acc0);
    acc1 = wmma4(a1, bm, acc1);
  }
  const float bb = bias[ncol];
#pragma unroll
  for (int r = 0; r < 8; ++r) {
    out[(size_t)(r + 8 * half)      * kD + ncol] = acc0[r] + bb;
    out[(size_t)(16 + r + 8 * half) * kD + ncol] = acc1[r] + bb;
  }
}

// ===========================================================================
// Kernel 2: context scores s[b,h,l] = scale * q[b,h,:] . Kcache[ci(b,l),h,:]
// One thread per row; 16x float4 (b128) loads, row = 256B fully consumed.
// grid = (B*H, L/256), block = 256
// ===========================================================================
__global__ __launch_bounds__(256)
void scores_kernel(const float* __restrict__ q, const float* __restrict__ kc,
                   const int* __restrict__ offsets,
                   const int* __restrict__ cind, float* __restrict__ s) {
  const int bh = blockIdx.x;
  const int b  = bh >> 5;
  const int hh = bh & 31;
  const int l  = blockIdx.y * 256 + threadIdx.x;  // 0..L-1
  const int ci = cind[offsets[b] + l];

  const float4* __restrict__ kp = (const float4*)(kc + (size_t)ci * kD + hh * kHD);
  const float4* __restrict__ qp = (const float4*)(q  + (size_t)b  * kD + hh * kHD);

  float acc = 0.f;
#pragma unroll
  for (int i = 0; i < kHD / 4; ++i) {
    float4 kv = kp[i];
    float4 qv = qp[i];
    acc += kv.x * qv.x + kv.y * qv.y + kv.z * qv.z + kv.w * qv.w;
  }
  s[(size_t)bh * (kL + 1) + l] = acc * kScale;
}

// ===========================================================================
// Kernel 3: append current-token score, softmax over (L+1) in place,
// and zero the output accumulator for kernel 4.
// grid = B*H, block = 256 (8 waves)
// ===========================================================================
__global__ __launch_bounds__(256)
void softmax_kernel(const float* __restrict__ q, const float* __restrict__ knew,
                    float* __restrict__ s, float* __restrict__ o) {
  const int bh   = blockIdx.x;
  const int b    = bh >> 5;
  const int hh   = bh & 31;
  const int lane = threadIdx.x & 31;
  const int wave = threadIdx.x >> 5;
  float* __restrict__ sp = s + (size_t)bh * (kL + 1);
  __shared__ float red[8];

  // current-token score (wave 0): q . k_new over 64 dims (v2f per lane)
  if (wave == 0) {
    v2f q2 = *(const v2f*)(q    + (size_t)b * kD + hh * kHD + lane * 2);
    v2f k2 = *(const v2f*)(knew + (size_t)b * kD + hh * kHD + lane * 2);
    float part = q2.x * k2.x + q2.y * k2.y;
#pragma unroll
    for (int d = 16; d > 0; d >>= 1) part += __shfl_xor(part, d, 32);
    if (lane == 0) sp[kL] = part * kScale;
  }
  __syncthreads();

  // block max
  float m = -3.402823466e38f;
  for (int i = threadIdx.x; i < kL + 1; i += 256) m = fmaxf(m, sp[i]);
#pragma unroll
  for (int d = 16; d > 0; d >>= 1) m = fmaxf(m, __shfl_xor(m, d, 32));
  if (lane == 0) red[wave] = m;
  __syncthreads();
  float gm = red[0];
#pragma unroll
  for (int i = 1; i < 8; ++i) gm = fmaxf(gm, red[i]);
  __syncthreads();

  // exp + block sum
  float ssum = 0.f;
  for (int i = threadIdx.x; i < kL + 1; i += 256) {
    float e = __expf(sp[i] - gm);
    sp[i] = e;
    ssum += e;
  }
#pragma unroll
  for (int d = 16; d > 0; d >>= 1) ssum += __shfl_xor(ssum, d, 32);
  if (lane == 0) red[wave] = ssum;
  __syncthreads();
  float gs = red[0];
#pragma unroll
  for (int i = 1; i < 8; ++i) gs += red[i];
  const float inv = 1.f / gs;

  for (int i = threadIdx.x; i < kL + 1; i += 256) sp[i] *= inv;

  // zero output accumulator (o read only in later kernels on this stream)
  if (threadIdx.x < kHD) o[(size_t)bh * kHD + threadIdx.x] = 0.f;
}

// ===========================================================================
// Kernel 4: o[b,h,:] += sum_l p[l] * Vcache[ci,h,:]  (+ p[L]*v_new on split 0)
// Each wave covers all 64 head dims (v2f/lane), 64 rows per wave.
// grid = (B*H, 4), block = 256 (8 waves) -> 4096 blocks streaming V.
// ===========================================================================
__global__ __launch_bounds__(256)
void pv_kernel(const float* __restrict__ s, const float* __restrict__ vc,
               const float* __restrict__ vnew, const int* __restrict__ offsets,
               const int* __restrict__ cind, float* __restrict__ o) {
  const int bh   = blockIdx.x;
  const int b    = bh >> 5;
  const int hh   = bh & 31;
  const int lane = threadIdx.x & 31;
  const int wave = threadIdx.x >> 5;
  const float* __restrict__ sp = s + (size_t)bh * (kL + 1);
  const int off = offsets[b];

  constexpr int kRows = kL / (4 * 8);  // 64 rows per wave
  const int l0 = (blockIdx.y * 8 + wave) * kRows;

  float ax = 0.f, ay = 0.f;
  for (int i = 0; i < kRows; ++i) {
    const int l = l0 + i;
    const float p = sp[l];
    const int ci = cind[off + l];
    v2f v2 = *(const v2f*)(vc + (size_t)ci * kD + hh * kHD + lane * 2);
    ax += p * v2.x;
    ay += p * v2.y;
  }
  if (blockIdx.y == 0 && wave == 0) {
    const float p = sp[kL];
    v2f v2 = *(const v2f*)(vnew + (size_t)b * kD + hh * kHD + lane * 2);
    ax += p * v2.x;
    ay += p * v2.y;
  }
  atomicAdd(&o[(size_t)bh * kHD + lane * 2],     ax);
  atomicAdd(&o[(size_t)bh * kHD + lane * 2 + 1], ay);
}

// ===========================================================================
// Kernel 5: out = o @ Wo + bo   (o viewed as [B, D] with col = h*64+d)
// grid = kD/16, block = 32
// ===========================================================================
__global__ __launch_bounds__(32)
void oproj_kernel(const float* __restrict__ o, const float* __restrict__ Wo,
                  const float* __restrict__ bo, float* __restrict__ out) {
  const int ntile = blockIdx.x;
  const int lane  = threadIdx.x & 31;
  const int m     = lane & 15;
  const int half  = lane >> 4;
  const int ncol  = ntile * 16 + m;

  v8f acc0 = {};
  v8f acc1 = {};
  for (int k0 = 0; k0 < kD; k0 += 4) {
    const int krow = k0 + 2 * half;
    v2f a0 = *(const v2f*)(o + (size_t)m        * kD + krow);
    v2f a1 = *(const v2f*)(o + (size_t)(16 + m) * kD + krow);
    v2f bm;
    bm.x = Wo[(size_t)krow       * kD + ncol];
    bm.y = Wo[(size_t)(krow + 1) * kD + ncol];
    acc0 = wmma4(a0, bm, acc0);
    acc1 = wmma4(a1, bm, acc1);
  }
  const float bb = bo[ncol];
#pragma unroll
  for (int r = 0; r < 8; ++r) {
    out[(size_t)(r + 8 * half)      * kD + ncol] = acc0[r] + bb;
    out[(size_t)(16 + r + 8 * half) * kD + ncol] = acc1[r] + bb;
  }
}

// ===========================================================================
extern "C" void kernel_launch(void* const* d_in, const int* in_sizes, int n_in,
                              void* d_out, int out_size, void* d_ws,
                              size_t ws_size, hipStream_t stream) {
  const float* h  = (const float*)d_in[0];
  const float* kc = (const float*)d_in[1];
  const float* vc = (const float*)d_in[2];
  const float* Wq = (const float*)d_in[3];
  const float* bq = (const float*)d_in[4];
  const float* Wk = (const float*)d_in[5];
  const float* bk = (const float*)d_in[6];
  const float* Wv = (const float*)d_in[7];
  const float* bv = (const float*)d_in[8];
  const float* Wo = (const float*)d_in[9];
  const float* bo = (const float*)d_in[10];
  const int* offsets = (const int*)d_in[11];
  const int* cind    = (const int*)d_in[12];
  float* out = (float*)d_out;

  float* ws = (float*)d_ws;
  float* q = ws;                                 // B*D
  float* k = q + (size_t)kB * kD;                // B*D
  float* v = k + (size_t)kB * kD;                // B*D
  float* s = v + (size_t)kB * kD;                // B*H*(L+1)
  float* o = s + (size_t)kB * kH * (kL + 1);     // B*H*64

  qkv_proj_kernel<<<dim3(kD / 16, 3), 32, 0, stream>>>(h, Wq, bq, Wk, bk, Wv,
                                                       bv, q, k, v);
  scores_kernel<<<dim3(kB * kH, kL / 256), 256, 0, stream>>>(q, kc, offsets,
                                                             cind, s);
  softmax_kernel<<<dim3(kB * kH), 256, 0, stream>>>(q, k, s, o);
  pv_kernel<<<dim3(kB * kH, 4), 256, 0, stream>>>(s, vc, v, offsets, cind, o);
  oproj_kernel<<<dim3(kD / 16), 32, 0, stream>>>(o, Wo, bo, out);
}